// WaveRNN_22694607192209
// MI455X (gfx1250) — compile-verified
//
#include <hip/hip_runtime.h>

// ---- problem constants (from reference) ----
#define NT    800
#define NZ    384
#define NX    384
#define NSRC  16
#define NREC  256
#define DT_   0.001f
#define DH_   10.0f

// ---- tiling ----
#define TS    32                       // 32x32 output tile per 256-thread block
#define LDSW  34                       // TS + 2 (halo)
#define NBLK  ((NZ / TS) * (NX / TS))  // 144 blocks per time step

#if defined(__AMDGCN__) && defined(__gfx1250__)
  #define ON_GFX1250 1
#else
  #define ON_GFX1250 0
#endif

#if ON_GFX1250 && __has_builtin(__builtin_amdgcn_global_load_async_to_lds_b32)
  #define HAVE_ASYNC_LDS 1
#else
  #define HAVE_ASYNC_LDS 0
#endif

typedef __attribute__((address_space(1))) int* global_i32_ptr;
typedef __attribute__((address_space(3))) int* lds_i32_ptr;

__device__ __forceinline__ void wait_async_zero() {
#if ON_GFX1250
  #if __has_builtin(__builtin_amdgcn_s_wait_asynccnt)
    __builtin_amdgcn_s_wait_asynccnt(0);
  #else
    asm volatile("s_wait_asynccnt 0" ::: "memory");
  #endif
#endif
}

__device__ __forceinline__ float agent_load(const float* p) {
    return __hip_atomic_load(p, __ATOMIC_RELAXED, __HIP_MEMORY_SCOPE_AGENT);
}
__device__ __forceinline__ void agent_store(float* p, float v) {
    __hip_atomic_store(p, v, __ATOMIC_RELAXED, __HIP_MEMORY_SCOPE_AGENT);
}

// One time step:  prev <- 2*cur - prev + c2 * lap(cur)   (in place on prev;
// only the center of prev is read, so this is race-free), then the LAST block
// to finish injects the 16 sources and gathers the 256 receivers.
__global__ __launch_bounds__(256)
void wave_step_kernel(const float* __restrict__ cur,
                      float* __restrict__ prev,
                      const float* __restrict__ c2,
                      const float* __restrict__ xt,     // x[t, :]  (NSRC floats)
                      const int*   __restrict__ src,    // (NSRC,2)
                      const int*   __restrict__ rec,    // (NREC,2)
                      float* __restrict__ out_t,        // d_out + t*NREC
                      unsigned int* counter) {
    __shared__ float tile[LDSW * LDSW];

    const int tid = threadIdx.x;            // 0..255
    const int bx  = blockIdx.x * TS;        // x origin of tile
    const int bz  = blockIdx.y * TS;        // z origin of tile

    // ---- stage (TS+2)x(TS+2) tile of cur into LDS, zero outside the domain ----
    for (int li = tid; li < LDSW * LDSW; li += 256) {
        int lz = li / LDSW;
        int lx = li - lz * LDSW;
        int gz = bz + lz - 1;
        int gx = bx + lx - 1;
        bool in = (gz >= 0) && (gz < NZ) && (gx >= 0) && (gx < NX);
#if HAVE_ASYNC_LDS
        if (in) {
            // async global -> LDS copy, tracked with ASYNCcnt (CDNA5 path)
            __builtin_amdgcn_global_load_async_to_lds_b32(
                (global_i32_ptr)(cur + gz * NX + gx),
                (lds_i32_ptr)(tile + li),
                0, 0);
        } else {
            tile[li] = 0.0f;   // disjoint address: no DS/ASYNC hazard
        }
#else
        tile[li] = in ? cur[gz * NX + gx] : 0.0f;
#endif
    }
#if HAVE_ASYNC_LDS
    wait_async_zero();
#endif
    __syncthreads();

    // ---- 5-point stencil: each thread updates 4 cells (z-strided) ----
    const int tx = tid & 31;
    const int tz = tid >> 5;                // 0..7
#pragma unroll
    for (int i = 0; i < 4; ++i) {
        int zz = tz + 8 * i;                // 0..31 within tile
        int lz = zz + 1, lx = tx + 1;
        float u   = tile[lz * LDSW + lx];
        float lap = tile[(lz - 1) * LDSW + lx] + tile[(lz + 1) * LDSW + lx]
                  + tile[lz * LDSW + lx - 1] + tile[lz * LDSW + lx + 1]
                  - 4.0f * u;
        int g = (bz + zz) * NX + (bx + tx);
        prev[g] = 2.0f * u - prev[g] + c2[g] * lap;
    }

    // ---- completion ticket: last block does source injection + recording ----
    __threadfence();                         // release this block's field stores
    __syncthreads();
    __shared__ unsigned int ticket;
    if (tid == 0)
        ticket = __hip_atomic_fetch_add(counter, 1u, __ATOMIC_ACQ_REL,
                                        __HIP_MEMORY_SCOPE_AGENT);
    __syncthreads();
    if (ticket == NBLK - 1) {
        __threadfence();                     // acquire all blocks' stores
        if (tid == 0) {
            // serial injection: deterministic even for duplicate source cells
            for (int s = 0; s < NSRC; ++s) {
                int idx = src[2 * s] * NX + src[2 * s + 1];
                agent_store(&prev[idx], agent_load(&prev[idx]) + xt[s]);
            }
        }
        __threadfence();
        __syncthreads();
        // 256 threads == NREC receivers; agent-scope load bypasses stale L0
        int idx = rec[2 * tid] * NX + rec[2 * tid + 1];
        out_t[tid] = agent_load(&prev[idx]);
        if (tid == 0)
            __hip_atomic_store(counter, 0u, __ATOMIC_RELAXED,
                               __HIP_MEMORY_SCOPE_AGENT);   // ready for next step
    }
}

// Zero both wavefields, precompute c2 = (c*DT/DH)^2, reset the ticket counter.
__global__ void wave_init_kernel(const float* __restrict__ c,
                                 float* __restrict__ hA,
                                 float* __restrict__ hB,
                                 float* __restrict__ c2,
                                 unsigned int* __restrict__ counter) {
    int i = blockIdx.x * blockDim.x + threadIdx.x;
    if (i < NZ * NX) {
        hA[i] = 0.0f;
        hB[i] = 0.0f;
        float s = c[i] * (DT_ / DH_);
        c2[i] = s * s;
    }
    if (i == 0) *counter = 0u;
}

extern "C" void kernel_launch(void* const* d_in, const int* in_sizes, int n_in,
                              void* d_out, int out_size, void* d_ws, size_t ws_size,
                              hipStream_t stream) {
    const float* x   = (const float*)d_in[0];   // (1, NT, NSRC) f32
    const float* c   = (const float*)d_in[1];   // (NZ, NX) f32
    const int*   src = (const int*)d_in[2];     // (NSRC, 2) i32
    const int*   rec = (const int*)d_in[3];     // (NREC, 2) i32
    float*       out = (float*)d_out;           // (1, NT, NREC) f32

    const int N = NZ * NX;
    float* hA = (float*)d_ws;                   // wavefield ping
    float* hB = hA + N;                         // wavefield pong
    float* c2 = hB + N;                         // (c*DT/DH)^2
    unsigned int* counter = (unsigned int*)(c2 + N);

    wave_init_kernel<<<(N + 255) / 256, 256, 0, stream>>>(c, hA, hB, c2, counter);

    dim3 grid(NX / TS, NZ / TS);                // 12 x 12 = 144 blocks
    float* cur  = hA;                           // h1
    float* prev = hB;                           // h2 (overwritten with h_next)
    for (int t = 0; t < NT; ++t) {
        wave_step_kernel<<<grid, 256, 0, stream>>>(
            cur, prev, c2, x + (size_t)t * NSRC, src, rec,
            out + (size_t)t * NREC, counter);
        float* tmp = cur; cur = prev; prev = tmp;   // (h_next, h1) -> (h1, h2)
    }

    (void)in_sizes; (void)n_in; (void)out_size; (void)ws_size;
}